// dc_layer_41094247088152
// MI455X (gfx1250) — compile-verified
//
#include <hip/hip_runtime.h>
#include <math.h>

#define FN  512
#define FNE (FN*FN)

typedef __attribute__((ext_vector_type(2))) float v2f;
typedef __attribute__((ext_vector_type(8))) float v8f;

__device__ __forceinline__ v8f wmma_f32_16x16x4(v2f a, v2f b, v8f c) {
  // 8 args: (neg_a, A, neg_b, B, c_mod, C, reuse_a, reuse_b)
  return __builtin_amdgcn_wmma_f32_16x16x4_f32(false, a, false, b, (short)0, c, false, false);
}

__device__ __forceinline__ void fill_tw(float2* tw, int tid, int nth) {
  for (int i = tid; i < FN; i += nth) {
    float ang = -6.28318530717958647692f * (float)i / (float)FN;
    float2 w; w.x = cosf(ang); w.y = sinf(ang);   // tw[j] = e^{-2*pi*i*j/512}
    tw[i] = w;
  }
}

// One 512-point FFT per wave32, data in LDS `row` (512 interleaved complex).
// Factorization: 512 = 16*32.  n = a + 16b (a in [0,16), b in [0,32)), k = d + 32c.
//   Stage1: Y[a][d]  = sum_b Z[a][b]  * w32^{b d}      (16x32x32 matmul, WMMA K-chunks of 4)
//   Twiddle:Y'[a][d] = Y[a][d] * w512^{a d}
//   Stage2: X[d+32c] = sum_a w16^{c a} * Y'[a][d]      (16x32x16 matmul)
// Inverse: conjugate all factors, scale by 1/512 per dimension.
__device__ void fft512_wave(float2* row, const float2* tw, int lane, int inverse, float scale) {
  const int hi = lane >> 4;       // 0 or 1
  const int lo = lane & 15;
  const float isgn = inverse ? -1.0f : 1.0f;
  const v8f vzero = {0,0,0,0,0,0,0,0};

  // ---- Stage 1 A operands: Z[a][b] = x[a + 16b]; A layout M=lo, K=r+2*hi per chunk
  v2f Ar[8], Ai[8];
#pragma unroll
  for (int q = 0; q < 8; ++q) {
#pragma unroll
    for (int r = 0; r < 2; ++r) {
      int b = 4*q + r + 2*hi;
      float2 zz = row[lo + 16*b];
      Ar[q][r] = zz.x;
      Ai[q][r] = zz.y;
    }
  }

  v8f Cr[2], Ci[2];
#pragma unroll
  for (int t = 0; t < 2; ++t) { Cr[t] = vzero; Ci[t] = vzero; }

#pragma unroll
  for (int q = 0; q < 8; ++q) {
#pragma unroll
    for (int t = 0; t < 2; ++t) {
      int d = lo + 16*t;
      v2f Br, Bi;
#pragma unroll
      for (int r = 0; r < 2; ++r) {
        int b = 4*q + r + 2*hi;
        float2 w = tw[(16*b*d) & (FN-1)];   // w32^{b d} = w512^{16 b d}
        Br[r] = w.x;
        Bi[r] = isgn * w.y;
      }
      v2f nBi = -Bi;
      // complex: Yr = Zr*Fr + Zi*(-Fi) ; Yi = Zr*Fi + Zi*Fr
      Cr[t] = wmma_f32_16x16x4(Ar[q], Br,  Cr[t]);
      Cr[t] = wmma_f32_16x16x4(Ai[q], nBi, Cr[t]);
      Ci[t] = wmma_f32_16x16x4(Ar[q], Bi,  Ci[t]);
      Ci[t] = wmma_f32_16x16x4(Ai[q], Br,  Ci[t]);
    }
  }

  __syncthreads();  // all reads of the input row complete

  // twiddle in the C/D layout (M = r8+8*hi, N = lo+16t), write Y'[a][d] -> row[a*32+d]
#pragma unroll
  for (int t = 0; t < 2; ++t) {
    int d = lo + 16*t;
#pragma unroll
    for (int r8 = 0; r8 < 8; ++r8) {
      int a = r8 + 8*hi;
      float2 w = tw[(a*d) & (FN-1)];
      float wr = w.x, wi = isgn * w.y;
      float yr = Cr[t][r8], yi = Ci[t][r8];
      float2 o; o.x = yr*wr - yi*wi; o.y = yr*wi + yi*wr;
      row[a*32 + d] = o;
    }
  }

  __syncthreads();  // Y' visible

  // ---- Stage 2: constants as A (M=c=lo, K=a), data as B (K=a, N=d)
  v8f Xr[2], Xi[2];
#pragma unroll
  for (int t = 0; t < 2; ++t) { Xr[t] = vzero; Xi[t] = vzero; }

#pragma unroll
  for (int q = 0; q < 4; ++q) {
    v2f Fr, Fi;
#pragma unroll
    for (int r = 0; r < 2; ++r) {
      int a = 4*q + r + 2*hi;
      float2 w = tw[(32*lo*a) & (FN-1)];    // w16^{c a} = w512^{32 c a}
      Fr[r] = w.x;
      Fi[r] = isgn * w.y;
    }
    v2f nFi = -Fi;
#pragma unroll
    for (int t = 0; t < 2; ++t) {
      int d = lo + 16*t;
      v2f Byr, Byi;
#pragma unroll
      for (int r = 0; r < 2; ++r) {
        int a = 4*q + r + 2*hi;
        float2 y = row[a*32 + d];
        Byr[r] = y.x; Byi[r] = y.y;
      }
      Xr[t] = wmma_f32_16x16x4(Fr,  Byr, Xr[t]);
      Xr[t] = wmma_f32_16x16x4(nFi, Byi, Xr[t]);
      Xi[t] = wmma_f32_16x16x4(Fr,  Byi, Xi[t]);
      Xi[t] = wmma_f32_16x16x4(Fi,  Byr, Xi[t]);
    }
  }

  __syncthreads();  // all reads of Y' complete

  // D element (M=c=r8+8*hi, N=d=lo+16t) -> output index k = d + 32c
#pragma unroll
  for (int t = 0; t < 2; ++t) {
    int d = lo + 16*t;
#pragma unroll
    for (int r8 = 0; r8 < 8; ++r8) {
      int c = r8 + 8*hi;
      float2 o; o.x = Xr[t][r8]*scale; o.y = Xi[t][r8]*scale;
      row[d + 32*c] = o;
    }
  }

  __syncthreads();  // results visible to caller
}

// ------------------- kernels -------------------

// Ssum[e] = sum_c sens[c][e]   (complex)
__global__ void k_coilsum(const float2* __restrict__ S, float2* __restrict__ Ssum, int C) {
  int e = blockIdx.x * 256 + threadIdx.x;
  float ax = 0.f, ay = 0.f;
  for (int c = 0; c < C; ++c) { float2 v = S[c*FNE + e]; ax += v.x; ay += v.y; }
  float2 o; o.x = ax; o.y = ay;
  Ssum[e] = o;
}

// rhs = nw + mu*z ; x=0, r=p=rhs ; Msh = fftshift(mask) ; partial ||rhs||^2
__global__ void k_init(const float2* __restrict__ nw, const float2* __restrict__ z,
                       const float* __restrict__ mask, const float* __restrict__ muP,
                       float2* __restrict__ x, float2* __restrict__ r, float2* __restrict__ p,
                       float* __restrict__ Msh, float* __restrict__ part) {
  __shared__ float red[256];
  const float mu = muP[0];
  float acc = 0.f;
  int base = blockIdx.x * 1024 + threadIdx.x;
#pragma unroll
  for (int k = 0; k < 4; ++k) {
    int e = base + k*256;
    float2 a = nw[e], b = z[e];
    float2 rr; rr.x = a.x + mu*b.x; rr.y = a.y + mu*b.y;
    r[e] = rr; p[e] = rr;
    float2 zz; zz.x = 0.f; zz.y = 0.f; x[e] = zz;
    int h = e >> 9, w = e & 511;
    Msh[e] = mask[(((h+256)&511) << 9) | ((w+256)&511)];
    acc += rr.x*rr.x + rr.y*rr.y;
  }
  red[threadIdx.x] = acc; __syncthreads();
  for (int s = 128; s > 0; s >>= 1) { if (threadIdx.x < s) red[threadIdx.x] += red[threadIdx.x+s]; __syncthreads(); }
  if (threadIdx.x == 0) part[blockIdx.x] = red[0];
}

__global__ void k_finish_rs0(const float* __restrict__ part, float* __restrict__ sc) {
  __shared__ float red[256];
  red[threadIdx.x] = part[threadIdx.x]; __syncthreads();
  for (int s = 128; s > 0; s >>= 1) { if (threadIdx.x < s) red[threadIdx.x] += red[threadIdx.x+s]; __syncthreads(); }
  if (threadIdx.x == 0) sc[0] = red[0];   // rsold
}

// row pass 1: u = Ssum*p (complex), forward row FFT -> Kb
__global__ void __launch_bounds__(128) k_rows1(const float2* __restrict__ Ssum,
                                               const float2* __restrict__ p,
                                               float2* __restrict__ Kb) {
  __shared__ float2 tw[FN];
  __shared__ float2 rows[4][FN];
  int tid = threadIdx.x, lane = tid & 31, wid = tid >> 5;
  fill_tw(tw, tid, 128);
  __syncthreads();
  int row = blockIdx.x * 4 + wid;
  const float2* s  = Ssum + (size_t)row * FN;
  const float2* pp = p    + (size_t)row * FN;
  for (int j = lane; j < FN; j += 32) {
    float2 a = s[j], b = pp[j];
    float2 u; u.x = a.x*b.x - a.y*b.y; u.y = a.x*b.y + a.y*b.x;
    rows[wid][j] = u;
  }
  __syncthreads();
  fft512_wave(rows[wid], tw, lane, 0, 1.0f);
  for (int j = lane; j < FN; j += 32) Kb[(size_t)row * FN + j] = rows[wid][j];
}

// col pass: forward col FFT, apply shifted mask, inverse col FFT (in place in Kb)
__global__ void __launch_bounds__(128) k_cols(float2* __restrict__ Kb, const float* __restrict__ Msh) {
  __shared__ float2 tw[FN];
  __shared__ float2 rows[4][FN];
  int tid = threadIdx.x, lane = tid & 31, wid = tid >> 5;
  fill_tw(tw, tid, 128);
  __syncthreads();
  int col = blockIdx.x * 4 + wid;
  for (int i = lane; i < FN; i += 32) rows[wid][i] = Kb[(size_t)i * FN + col];
  __syncthreads();
  fft512_wave(rows[wid], tw, lane, 0, 1.0f);
  for (int i = lane; i < FN; i += 32) {
    float m = Msh[(size_t)i * FN + col];
    float2 v = rows[wid][i]; v.x *= m; v.y *= m; rows[wid][i] = v;
  }
  __syncthreads();
  fft512_wave(rows[wid], tw, lane, 1, 1.0f/512.0f);
  for (int i = lane; i < FN; i += 32) Kb[(size_t)i * FN + col] = rows[wid][i];
}

// row pass 2: inverse row FFT; Ap = conj(Ssum)*img + mu*p; partial Re<p,Ap>
__global__ void __launch_bounds__(128) k_rows2(const float2* __restrict__ Kb,
                                               const float2* __restrict__ Ssum,
                                               const float2* __restrict__ p,
                                               const float* __restrict__ muP,
                                               float2* __restrict__ Ap,
                                               float* __restrict__ part) {
  __shared__ float2 tw[FN];
  __shared__ float2 rows[4][FN];
  __shared__ float red[128];
  int tid = threadIdx.x, lane = tid & 31, wid = tid >> 5;
  fill_tw(tw, tid, 128);
  __syncthreads();
  int row = blockIdx.x * 4 + wid;
  for (int j = lane; j < FN; j += 32) rows[wid][j] = Kb[(size_t)row * FN + j];
  __syncthreads();
  fft512_wave(rows[wid], tw, lane, 1, 1.0f/512.0f);
  const float mu = muP[0];
  float acc = 0.f;
  for (int j = lane; j < FN; j += 32) {
    float2 s = Ssum[(size_t)row*FN + j];
    float2 g = rows[wid][j];
    float2 pv = p[(size_t)row*FN + j];
    float2 ap;
    ap.x = s.x*g.x + s.y*g.y + mu*pv.x;   // conj(s)*g
    ap.y = s.x*g.y - s.y*g.x + mu*pv.y;
    Ap[(size_t)row*FN + j] = ap;
    acc += pv.x*ap.x + pv.y*ap.y;
  }
  red[tid] = acc; __syncthreads();
  for (int s = 64; s > 0; s >>= 1) { if (tid < s) red[tid] += red[tid+s]; __syncthreads(); }
  if (tid == 0) part[blockIdx.x] = red[0];
}

__global__ void k_finish_alpha(const float* __restrict__ part, float* __restrict__ sc) {
  __shared__ float red[128];
  red[threadIdx.x] = part[threadIdx.x]; __syncthreads();
  for (int s = 64; s > 0; s >>= 1) { if (threadIdx.x < s) red[threadIdx.x] += red[threadIdx.x+s]; __syncthreads(); }
  if (threadIdx.x == 0) sc[1] = sc[0] / red[0];   // alpha = rsold / pAp
}

// x += alpha p ; r -= alpha Ap ; partial ||r||^2
__global__ void k_update1(float2* __restrict__ x, float2* __restrict__ r,
                          const float2* __restrict__ p, const float2* __restrict__ Ap,
                          const float* __restrict__ sc, float* __restrict__ part) {
  __shared__ float red[256];
  const float alpha = sc[1];
  float acc = 0.f;
  int base = blockIdx.x * 1024 + threadIdx.x;
#pragma unroll
  for (int k = 0; k < 4; ++k) {
    int e = base + k*256;
    float2 pv = p[e], av = Ap[e];
    float2 xv = x[e]; xv.x += alpha*pv.x; xv.y += alpha*pv.y; x[e] = xv;
    float2 rv = r[e]; rv.x -= alpha*av.x; rv.y -= alpha*av.y; r[e] = rv;
    acc += rv.x*rv.x + rv.y*rv.y;
  }
  red[threadIdx.x] = acc; __syncthreads();
  for (int s = 128; s > 0; s >>= 1) { if (threadIdx.x < s) red[threadIdx.x] += red[threadIdx.x+s]; __syncthreads(); }
  if (threadIdx.x == 0) part[blockIdx.x] = red[0];
}

__global__ void k_finish_beta(const float* __restrict__ part, float* __restrict__ sc) {
  __shared__ float red[256];
  red[threadIdx.x] = part[threadIdx.x]; __syncthreads();
  for (int s = 128; s > 0; s >>= 1) { if (threadIdx.x < s) red[threadIdx.x] += red[threadIdx.x+s]; __syncthreads(); }
  if (threadIdx.x == 0) {
    float rsnew = red[0];
    sc[2] = rsnew / sc[0];   // beta
    sc[0] = rsnew;           // rsold = rsnew
  }
}

__global__ void k_update2(float2* __restrict__ p, const float2* __restrict__ r,
                          const float* __restrict__ sc) {
  const float beta = sc[2];
  int e = blockIdx.x * 256 + threadIdx.x;
  float2 rv = r[e], pv = p[e];
  float2 o; o.x = rv.x + beta*pv.x; o.y = rv.y + beta*pv.y;
  p[e] = o;
}

__global__ void k_output(const float* __restrict__ x, float* __restrict__ out) {
  int i = blockIdx.x * 256 + threadIdx.x;
  out[i] = x[i];
}

// ------------------- launch -------------------

extern "C" void kernel_launch(void* const* d_in, const int* in_sizes, int n_in,
                              void* d_out, int out_size, void* d_ws, size_t ws_size,
                              hipStream_t stream) {
  const float2* S    = (const float2*)d_in[0];   // [C,512,512,2]
  const float*  mask = (const float*)d_in[1];    // [512,512]
  const float2* nw   = (const float2*)d_in[2];   // [512,512,2]
  const float2* z    = (const float2*)d_in[3];   // [512,512,2]
  const float*  mu   = (const float*)d_in[4];    // [1]
  const int C = in_sizes[0] / (FNE * 2);

  float* ws = (float*)d_ws;
  float2* Ssum = (float2*)(ws + 0);          // 2*FNE floats
  float2* Kb   = (float2*)(ws + 2*(size_t)FNE);
  float2* Ap   = (float2*)(ws + 4*(size_t)FNE);
  float2* x    = (float2*)(ws + 6*(size_t)FNE);
  float2* r    = (float2*)(ws + 8*(size_t)FNE);
  float2* p    = (float2*)(ws + 10*(size_t)FNE);
  float*  Msh  = ws + 12*(size_t)FNE;        // FNE floats
  float*  part = ws + 13*(size_t)FNE;        // 1024 floats
  float*  sc   = ws + 13*(size_t)FNE + 1024; // scalars: [0]=rsold [1]=alpha [2]=beta

  k_coilsum<<<FNE/256, 256, 0, stream>>>(S, Ssum, C);
  k_init<<<256, 256, 0, stream>>>(nw, z, mask, mu, x, r, p, Msh, part);
  k_finish_rs0<<<1, 256, 0, stream>>>(part, sc);

  for (int it = 0; it < 10; ++it) {
    k_rows1<<<128, 128, 0, stream>>>(Ssum, p, Kb);
    k_cols<<<128, 128, 0, stream>>>(Kb, Msh);
    k_rows2<<<128, 128, 0, stream>>>(Kb, Ssum, p, mu, Ap, part);
    k_finish_alpha<<<1, 128, 0, stream>>>(part, sc);
    k_update1<<<256, 256, 0, stream>>>(x, r, p, Ap, sc, part);
    k_finish_beta<<<1, 256, 0, stream>>>(part, sc);
    k_update2<<<FNE/256, 256, 0, stream>>>(p, r, sc);
  }

  k_output<<<2*FNE/256, 256, 0, stream>>>((const float*)x, (float*)d_out);
}